// Preprocessor_39298950758647
// MI455X (gfx1250) — compile-verified
//
#include <hip/hip_runtime.h>
#include <hip/hip_bf16.h>

#define H_IMG 4096
#define W_IMG 4096
#define T_TILES 8
#define BINS 256
#define TH_T (H_IMG / T_TILES)  /* 512 */
#define TW_T (W_IMG / T_TILES)  /* 512 */
#define AREA (TH_T * TW_T)      /* 262144 */
#define CLIP_CNT 40960          /* max(int(40.0*AREA/256),1) */
#define NCH 4
#define NPIX (H_IMG * W_IMG)

typedef __attribute__((ext_vector_type(2))) float v2f;
typedef __attribute__((ext_vector_type(8))) float v8f;

// ---- helpers (must be bit-identical between histogram and apply passes) ----
__device__ __forceinline__ float norm1(float x, float vmin, float rng, int cond) {
  // masked_normalize with lo=0, hi=1; mask = x > 0
  if (x > 0.0f) return cond ? (x - vmin) / rng : 0.0f;
  return x;
}
__device__ __forceinline__ int quant255(float y) {
  float f = floorf(y * 255.0f);
  f = fminf(fmaxf(f, 0.0f), 255.0f);
  return (int)f;
}

// ---------------- K0: init atomic reduction cells ----------------
__global__ void k_init(unsigned* smin1, unsigned* smax1, unsigned* smin2, unsigned* smax2) {
  int t = threadIdx.x;
  if (t < NCH) {
    smin1[t] = 0x7F800000u; smax1[t] = 0u;  // +inf / "empty"
    smin2[t] = 0x7F800000u; smax2[t] = 0u;
  }
}

// ---------------- K1: per-channel masked min/max (mask = x>0) ----------------
__global__ void k_minmax1(const float* __restrict__ X, unsigned* __restrict__ smin,
                          unsigned* __restrict__ smax) {
  const int ch = blockIdx.y;
  const float4* p = (const float4*)(X + (size_t)ch * NPIX);
  const int nvec = NPIX / 4;
  float lmin = __builtin_inff(), lmax = 0.0f;
  for (int i = blockIdx.x * blockDim.x + threadIdx.x; i < nvec; i += gridDim.x * blockDim.x) {
    float4 v = p[i];
    if (v.x > 0.0f) { lmin = fminf(lmin, v.x); lmax = fmaxf(lmax, v.x); }
    if (v.y > 0.0f) { lmin = fminf(lmin, v.y); lmax = fmaxf(lmax, v.y); }
    if (v.z > 0.0f) { lmin = fminf(lmin, v.z); lmax = fmaxf(lmax, v.z); }
    if (v.w > 0.0f) { lmin = fminf(lmin, v.w); lmax = fmaxf(lmax, v.w); }
  }
  __shared__ float rmin[256], rmax[256];
  const int tid = threadIdx.x;
  rmin[tid] = lmin; rmax[tid] = lmax; __syncthreads();
  for (int s = 128; s > 0; s >>= 1) {
    if (tid < s) { rmin[tid] = fminf(rmin[tid], rmin[tid + s]); rmax[tid] = fmaxf(rmax[tid], rmax[tid + s]); }
    __syncthreads();
  }
  if (tid == 0) {
    atomicMin(&smin[ch], __float_as_uint(rmin[0]));
    atomicMax(&smax[ch], __float_as_uint(rmax[0]));  // monotone for non-negative floats
  }
}

// ---------------- K2: per-(channel,tile) 256-bin histogram ----------------
__global__ void k_hist(const float* __restrict__ X, const unsigned* __restrict__ smin,
                       const unsigned* __restrict__ smax, unsigned* __restrict__ hist) {
  const int ch = blockIdx.y, tile = blockIdx.x;
  const int tr = tile >> 3, tc = tile & 7;
  const float vmin = __uint_as_float(smin[ch]);
  const unsigned mxb = smax[ch];
  const int cond = (mxb != 0u);  // any(mask) && vmax>0 (masked values strictly positive)
  const float rng = fmaxf(__uint_as_float(mxb) - vmin, 1e-12f);

  __shared__ unsigned lh[BINS];
  for (int b = threadIdx.x; b < BINS; b += blockDim.x) lh[b] = 0u;
  __syncthreads();

  const float* base = X + (size_t)ch * NPIX + (size_t)tr * TH_T * W_IMG + (size_t)tc * TW_T;
  const int iters = (TH_T * TW_T) / (256 * 4);  // 256
  for (int it = 0; it < iters; ++it) {
    const int flat = (it * 256 + threadIdx.x) * 4;
    const int r = flat / TW_T, c = flat % TW_T;
    const float* addr = base + (size_t)r * W_IMG + c;
    __builtin_prefetch(addr + 2 * (size_t)W_IMG, 0, 1);  // global_prefetch_b8
    const float4 v = *(const float4*)addr;
    atomicAdd(&lh[quant255(norm1(v.x, vmin, rng, cond))], 1u);
    atomicAdd(&lh[quant255(norm1(v.y, vmin, rng, cond))], 1u);
    atomicAdd(&lh[quant255(norm1(v.z, vmin, rng, cond))], 1u);
    atomicAdd(&lh[quant255(norm1(v.w, vmin, rng, cond))], 1u);
  }
  __syncthreads();
  unsigned* hdst = hist + ((size_t)ch * 64 + tile) * BINS;
  for (int b = threadIdx.x; b < BINS; b += blockDim.x) hdst[b] = lh[b];  // one block per tile: plain store
}

// ---------------- K3: per-channel nonzero count + mean of nonzero codes ----------------
__global__ void k_stats(const unsigned* __restrict__ hist, int* __restrict__ mean_nz,
                        int* __restrict__ nzflag) {
  const int ch = blockIdx.x;
  const unsigned* h = hist + (size_t)ch * 64 * BINS;
  double s = 0.0; long long z = 0;
  for (int i = threadIdx.x; i < 64 * BINS; i += blockDim.x) {
    const unsigned c = h[i];
    const int b = i & (BINS - 1);
    if (b == 0) z += (long long)c; else s += (double)c * (double)b;
  }
  __shared__ double sd[256]; __shared__ long long sz[256];
  sd[threadIdx.x] = s; sz[threadIdx.x] = z; __syncthreads();
  for (int st = 128; st > 0; st >>= 1) {
    if (threadIdx.x < st) { sd[threadIdx.x] += sd[threadIdx.x + st]; sz[threadIdx.x] += sz[threadIdx.x + st]; }
    __syncthreads();
  }
  if (threadIdx.x == 0) {
    const long long nz = (long long)NPIX - sz[0];
    const long long den = nz > 0 ? nz : 1;
    mean_nz[ch] = (int)floor(sd[0] / (double)den);
    nzflag[ch] = nz > 0 ? 1 : 0;
  }
}

// ---------------- K4: clip/redistribute + WMMA prefix-sum -> LUT ----------------
// One wave (32 threads) per (channel, tile). Inclusive scan of 256 bins:
// view bins as 16x16 matrix H (row r = bins 16r..16r+15); within-row scan is
// S = H x U with U upper-triangular ones, computed as 4 chained
// V_WMMA_F32_16X16X4_F32 (K=16 split into 4 chunks); then add row offsets.
__global__ void k_lut(const unsigned* __restrict__ hist, const int* __restrict__ mean_nz,
                      const int* __restrict__ nzflag, float* __restrict__ lut) {
  const int ch = blockIdx.y, tile = blockIdx.x;
  const int lane = threadIdx.x;  // 0..31, exactly one wave32
  __shared__ float sh[BINS];
  __shared__ float scn[BINS];
  __shared__ float exs[32];

  const unsigned* hsrc = hist + ((size_t)ch * 64 + tile) * BINS;
  for (int i = 0; i < 8; ++i) sh[lane * 8 + i] = (float)hsrc[lane * 8 + i];
  __syncthreads();

  const int mean = mean_nz[ch], nzf = nzflag[ch];
  if (lane == 0 && nzf && mean != 0) {  // filled = where(xb==0 & nz>0, mean, xb)
    sh[mean] += sh[0];
    sh[0] = 0.0f;
  }
  __syncthreads();

  const float clf = (float)CLIP_CNT;
  float ex = 0.0f;
  for (int i = 0; i < 8; ++i) ex += fmaxf(sh[lane * 8 + i] - clf, 0.0f);
  exs[lane] = ex; __syncthreads();
  if (lane == 0) { float t = 0.0f; for (int i = 0; i < 32; ++i) t += exs[i]; exs[0] = t; }
  __syncthreads();
  const float excess = exs[0];
  const float batch = floorf(excess / 256.0f);
  const float residual = excess - batch * 256.0f;
  const float step = fmaxf(floorf(256.0f / fmaxf(residual, 1.0f)), 1.0f);
  for (int i = 0; i < 8; ++i) {
    const int b = lane * 8 + i;
    const float bf = (float)b;
    const float v = fminf(sh[b], clf);
    const float ra = ((fmodf(bf, step) == 0.0f) && (floorf(bf / step) < residual)) ? 1.0f : 0.0f;
    sh[b] = v + batch + ra;
  }
  __syncthreads();

  // --- WMMA within-row scan: D = A(16x4 chunks of H) x B(4x16 triangular) + C ---
  const int hi = lane >> 4;   // lane half selects K pair (A) / K pair (B)
  const int m = lane & 15;    // A row / B,D column
  v8f acc = {};
  for (int kk = 0; kk < 4; ++kk) {
    const int k0 = 4 * kk + 2 * hi;
    v2f a, b;
    a.x = sh[m * 16 + k0];
    a.y = sh[m * 16 + k0 + 1];
    b.x = (k0 <= m) ? 1.0f : 0.0f;       // U[k][n] = (k <= n), n = m
    b.y = (k0 + 1 <= m) ? 1.0f : 0.0f;
    acc = __builtin_amdgcn_wmma_f32_16x16x4_f32(
        /*neg_a=*/false, a, /*neg_b=*/false, b,
        /*c_mod=*/(short)0, acc, /*reuse_a=*/false, /*reuse_b=*/false);
  }
  // D layout: VGPR i holds row (i + 8*hi), column m
  for (int i = 0; i < 8; ++i) scn[(i + 8 * hi) * 16 + m] = acc[i];
  __syncthreads();

  float* ldst = lut + ((size_t)ch * 64 + tile) * BINS;
  for (int i = 0; i < 8; ++i) {
    const int bidx = lane * 8 + i;
    const int r = bidx >> 4, k = bidx & 15;
    float off = 0.0f;
    for (int q = 0; q < r; ++q) off += scn[q * 16 + 15];  // exclusive row-prefix
    const float cdf = scn[r * 16 + k] + off;
    const float l = rintf(cdf * (255.0f / (float)AREA));  // jnp.round = RNE
    ldst[bidx] = fminf(fmaxf(l, 0.0f), 255.0f);
  }
}

// ---------------- K5: bilinear LUT apply + fused masked min/max of result ----------------
__global__ void k_apply(const float* __restrict__ X, const float* __restrict__ lut,
                        const unsigned* __restrict__ smin1, const unsigned* __restrict__ smax1,
                        const int* __restrict__ mean_nz, const int* __restrict__ nzflag,
                        float* __restrict__ out, unsigned* __restrict__ smin2,
                        unsigned* __restrict__ smax2) {
  const int ch = blockIdx.y;
  const float vmin = __uint_as_float(smin1[ch]);
  const unsigned mxb = smax1[ch];
  const int cond = (mxb != 0u);
  const float rng = fmaxf(__uint_as_float(mxb) - vmin, 1e-12f);
  const int mean = mean_nz[ch], nzf = nzflag[ch];
  const float* xch = X + (size_t)ch * NPIX;
  float* och = out + (size_t)ch * NPIX;
  const float* lch = lut + (size_t)ch * 64 * BINS;

  const int gid = blockIdx.x * blockDim.x + threadIdx.x;  // 0..NPIX/4-1
  const int flat = gid * 4;
  const int row = flat >> 12;      // /4096
  const int col = flat & 4095;

  const float fy = ((float)row + 0.5f) / (float)TH_T - 0.5f;
  const float fyf = floorf(fy);
  int y0 = (int)fyf; y0 = min(max(y0, 0), T_TILES - 1);
  const int y1 = min(y0 + 1, T_TILES - 1);
  const float wy = fy - fyf;

  const float4 v = *(const float4*)(xch + flat);
  float xin[4] = {v.x, v.y, v.z, v.w};
  float eo[4];
  float lmin = __builtin_inff(), lmax = 0.0f;

#pragma unroll
  for (int j = 0; j < 4; ++j) {
    const float fx = ((float)(col + j) + 0.5f) / (float)TW_T - 0.5f;
    const float fxf = floorf(fx);
    int x0 = (int)fxf; x0 = min(max(x0, 0), T_TILES - 1);
    const int x1 = min(x0 + 1, T_TILES - 1);
    const float wx = fx - fxf;

    const int xb = quant255(norm1(xin[j], vmin, rng, cond));
    const bool zero = (xb == 0);
    const int fil = (zero && nzf) ? mean : xb;

    const float v00 = lch[((y0 * 8 + x0) << 8) + fil];
    const float v01 = lch[((y0 * 8 + x1) << 8) + fil];
    const float v10 = lch[((y1 * 8 + x0) << 8) + fil];
    const float v11 = lch[((y1 * 8 + x1) << 8) + fil];
    float eq = v00 * (1.0f - wy) * (1.0f - wx) + v01 * (1.0f - wy) * wx +
               v10 * wy * (1.0f - wx) + v11 * wy * wx;
    if (zero) eq = 0.0f;
    const float e = eq / 255.0f;
    eo[j] = e;
    if (xin[j] > 0.0f) { lmin = fminf(lmin, e); lmax = fmaxf(lmax, e); }
  }
  *(float4*)(och + flat) = make_float4(eo[0], eo[1], eo[2], eo[3]);

  __shared__ float rmin[256], rmax[256];
  const int tid = threadIdx.x;
  rmin[tid] = lmin; rmax[tid] = lmax; __syncthreads();
  for (int s = 128; s > 0; s >>= 1) {
    if (tid < s) { rmin[tid] = fminf(rmin[tid], rmin[tid + s]); rmax[tid] = fmaxf(rmax[tid], rmax[tid + s]); }
    __syncthreads();
  }
  if (tid == 0) {
    atomicMin(&smin2[ch], __float_as_uint(rmin[0]));
    atomicMax(&smax2[ch], __float_as_uint(rmax[0]));
  }
}

// ---------------- K6: final masked_normalize to [1/255, 1] ----------------
__global__ void k_final(const float* __restrict__ X, float* __restrict__ out,
                        const unsigned* __restrict__ smin2, const unsigned* __restrict__ smax2) {
  const int ch = blockIdx.y;
  const unsigned mxb = smax2[ch];
  const int cond = (mxb != 0u);  // any(mask) && vmax>0 collapse to "max bits nonzero"
  const float vmin = __uint_as_float(smin2[ch]);
  const float rng = fmaxf(__uint_as_float(mxb) - vmin, 1e-12f);
  const float lo = 1.0f / 255.0f;
  const float span = 1.0f - lo;

  const float* xch = X + (size_t)ch * NPIX;
  float* och = out + (size_t)ch * NPIX;
  const int gid = blockIdx.x * blockDim.x + threadIdx.x;
  const int flat = gid * 4;
  const float4 xv = *(const float4*)(xch + flat);
  float4 ev = *(const float4*)(och + flat);
  float xs[4] = {xv.x, xv.y, xv.z, xv.w};
  float es[4] = {ev.x, ev.y, ev.z, ev.w};
#pragma unroll
  for (int j = 0; j < 4; ++j) {
    if (xs[j] > 0.0f)
      es[j] = cond ? ((es[j] - vmin) / rng * span + lo) : lo;
    // else: keep e (reference returns x-arg of masked_normalize, which is e here)
  }
  *(float4*)(och + flat) = make_float4(es[0], es[1], es[2], es[3]);
}

extern "C" void kernel_launch(void* const* d_in, const int* in_sizes, int n_in,
                              void* d_out, int out_size, void* d_ws, size_t ws_size,
                              hipStream_t stream) {
  const float* X = (const float*)d_in[0];
  float* out = (float*)d_out;
  char* ws = (char*)d_ws;

  unsigned* smin1 = (unsigned*)(ws + 0);
  unsigned* smax1 = (unsigned*)(ws + 64);
  unsigned* smin2 = (unsigned*)(ws + 128);
  unsigned* smax2 = (unsigned*)(ws + 192);
  int* mean_nz    = (int*)(ws + 256);
  int* nzflag     = (int*)(ws + 320);
  unsigned* hist  = (unsigned*)(ws + 512);                      // 4*64*256*4 = 256 KiB
  float* lut      = (float*)(ws + 512 + (size_t)NCH * 64 * BINS * 4);  // 256 KiB

  k_init<<<dim3(1), dim3(32), 0, stream>>>(smin1, smax1, smin2, smax2);
  k_minmax1<<<dim3(1024, NCH), dim3(256), 0, stream>>>(X, smin1, smax1);
  k_hist<<<dim3(64, NCH), dim3(256), 0, stream>>>(X, smin1, smax1, hist);
  k_stats<<<dim3(NCH), dim3(256), 0, stream>>>(hist, mean_nz, nzflag);
  k_lut<<<dim3(64, NCH), dim3(32), 0, stream>>>(hist, mean_nz, nzflag, lut);
  k_apply<<<dim3(NPIX / (256 * 4), NCH), dim3(256), 0, stream>>>(
      X, lut, smin1, smax1, mean_nz, nzflag, out, smin2, smax2);
  k_final<<<dim3(NPIX / (256 * 4), NCH), dim3(256), 0, stream>>>(X, out, smin2, smax2);
}